// MLPConcatHead_origin_87393994539112
// MI455X (gfx1250) — compile-verified
//
#include <hip/hip_runtime.h>
#include <hip/hip_bf16.h>

// ---------------------------------------------------------------------------
// MLPConcatHead for MI455X (gfx1250, wave32, WMMA).
// - All GEMMs use V_WMMA_F32_16X16X4_F32 (full f32; matmul is <1% of work,
//   so we keep precision instead of the bf16 K=32 path).
// - The O(N^2*Hd) pair pass is fused; LN stats decompose so the per-pair
//   variance comes from a per-tile WMMA cross-product GEMM:
//     var_ij = mean(A_i^2) + 2*mean(A_i.B_j) + mean(B_j^2)
// - LDS pitch 260 (= 4 mod 64): conflict-free banking AND 16B alignment so
//   the hot mish loop uses ds_load_b128 (4x fewer LDS ops -> pure VALU bound).
// ---------------------------------------------------------------------------

typedef float v2f __attribute__((ext_vector_type(2)));
typedef float v8f __attribute__((ext_vector_type(8)));

#define WMMA_F32(a, b, c) \
  __builtin_amdgcn_wmma_f32_16x16x4_f32(false, (a), false, (b), (short)0, (c), false, false)

constexpr int N_   = 768;   // rows
constexpr int D_   = 256;   // input dim
constexpr int HD_  = 256;   // hidden dim
constexpr int P_   = 128;   // pair feature dim
constexpr int PAD  = 260;   // pitch for 256-wide tiles (%64==4, %4==0)
constexpr int PADH = 132;   // pitch for 128-wide tiles (%64==4, %4==0)

__device__ __forceinline__ float mishf(float x) {
  // mish(x) = x * tanh(softplus(x)) = x * (e^2+2e)/(e^2+2e+2), e = exp(x)
  if (x > 15.0f) return x;                 // tanh(softplus(x)) ~ 1
  float e = __expf(x);
  float n = e * (e + 2.0f);
  return x * n / (n + 2.0f);
}
__device__ __forceinline__ float hsum4(float4 v) { return (v.x + v.y) + (v.z + v.w); }
__device__ __forceinline__ float4 sub4(float4 v, float s) {
  return make_float4(v.x - s, v.y - s, v.z - s, v.w - s);
}
__device__ __forceinline__ float dot4(float4 a, float4 b) {
  return a.x * b.x + a.y * b.y + a.z * b.z + a.w * b.w;
}

// ---------------------------------------------------------------------------
// K1: H1 = mish(LN(h_fused @ W0 + b0))          (768x256) x (256x256)
// One block = 16 rows. 8 waves, each wave computes 16x32 (two 16x16 tiles).
// ---------------------------------------------------------------------------
__global__ __launch_bounds__(256) void k_gemm1_ln_mish(
    const float* __restrict__ hf, const float* __restrict__ W0,
    const float* __restrict__ b0, const float* __restrict__ g0,
    const float* __restrict__ be0, float* __restrict__ H1) {
  __shared__ __align__(16) float sIn[16 * PAD];
  __shared__ __align__(16) float sOut[16 * PAD];
  const int tid  = threadIdx.x;
  const int lane = tid & 31;
  const int wave = tid >> 5;
  const int row0 = blockIdx.x * 16;

  {  // stage 16x256 block, b128 coalesced
    const int cc = (tid & 63) * 4, r0 = tid >> 6;
    for (int q = 0; q < 4; ++q) {
      int rr = r0 + q * 4;
      *(float4*)&sIn[rr * PAD + cc] = *(const float4*)&hf[(row0 + rr) * D_ + cc];
    }
  }
  __syncthreads();

  const int m  = lane & 15;          // A row / B col within fragment
  const int ks = (lane >> 4) * 2;    // K sub-select per half-wave
  const int col0 = wave * 32;
  v8f c0 = {}; v8f c1 = {};
  for (int k0 = 0; k0 < D_; k0 += 4) {
    v2f a;
    a.x = sIn[m * PAD + k0 + ks];
    a.y = sIn[m * PAD + k0 + ks + 1];
    const float* wp = &W0[(k0 + ks) * HD_ + m];
    v2f bA; bA.x = wp[col0];      bA.y = wp[HD_ + col0];
    v2f bB; bB.x = wp[col0 + 16]; bB.y = wp[HD_ + col0 + 16];
    c0 = WMMA_F32(a, bA, c0);
    c1 = WMMA_F32(a, bB, c1);
  }
  const int hi = (lane >> 4) * 8;
  for (int r = 0; r < 8; ++r) {
    int rr = r + hi;
    sOut[rr * PAD + col0 + m]      = c0[r] + b0[col0 + m];
    sOut[rr * PAD + col0 + 16 + m] = c1[r] + b0[col0 + 16 + m];
  }
  __syncthreads();

  // LayerNorm + mish: row r handled by 16 threads; values held in registers.
  const int r  = tid >> 4;
  const int cg = tid & 15;
  const float4* prow = (const float4*)&sOut[r * PAD + cg * 16];
  float4 x0 = prow[0], x1 = prow[1], x2 = prow[2], x3 = prow[3];
  float s = hsum4(x0) + hsum4(x1) + hsum4(x2) + hsum4(x3);
  for (int o = 8; o >= 1; o >>= 1) s += __shfl_xor(s, o, 16);
  float mean = s * (1.0f / 256.0f);
  float4 dd[4] = {sub4(x0, mean), sub4(x1, mean), sub4(x2, mean), sub4(x3, mean)};
  float vs = dot4(dd[0], dd[0]) + dot4(dd[1], dd[1]) +
             dot4(dd[2], dd[2]) + dot4(dd[3], dd[3]);
  for (int o = 8; o >= 1; o >>= 1) vs += __shfl_xor(vs, o, 16);
  float rstd = rsqrtf(vs * (1.0f / 256.0f) + 1e-5f);
#pragma unroll
  for (int q = 0; q < 4; ++q) {
    float4 g = ((const float4*)g0)[cg * 4 + q];
    float4 b = ((const float4*)be0)[cg * 4 + q];
    float4 o4;
    o4.x = mishf(dd[q].x * rstd * g.x + b.x);
    o4.y = mishf(dd[q].y * rstd * g.y + b.y);
    o4.z = mishf(dd[q].z * rstd * g.z + b.z);
    o4.w = mishf(dd[q].w * rstd * g.w + b.w);
    ((float4*)&H1[(row0 + r) * HD_])[cg * 4 + q] = o4;
  }
}

// ---------------------------------------------------------------------------
// K2: H = H1 @ W1 + b1                          (768x256) x (256x128)
// One block = 16 rows; 8 waves, each one 16x16 tile (8*16 = 128 cols).
// ---------------------------------------------------------------------------
__global__ __launch_bounds__(256) void k_gemm2(
    const float* __restrict__ H1, const float* __restrict__ W1,
    const float* __restrict__ b1, float* __restrict__ H) {
  __shared__ __align__(16) float sIn[16 * PAD];
  const int tid  = threadIdx.x;
  const int lane = tid & 31;
  const int wave = tid >> 5;
  const int row0 = blockIdx.x * 16;

  {
    const int cc = (tid & 63) * 4, r0 = tid >> 6;
    for (int q = 0; q < 4; ++q) {
      int rr = r0 + q * 4;
      *(float4*)&sIn[rr * PAD + cc] = *(const float4*)&H1[(row0 + rr) * HD_ + cc];
    }
  }
  __syncthreads();

  const int m  = lane & 15;
  const int ks = (lane >> 4) * 2;
  const int col0 = wave * 16;
  v8f c = {};
  for (int k0 = 0; k0 < HD_; k0 += 4) {
    v2f a;
    a.x = sIn[m * PAD + k0 + ks];
    a.y = sIn[m * PAD + k0 + ks + 1];
    v2f b;
    b.x = W1[(k0 + ks) * P_ + col0 + m];
    b.y = W1[(k0 + ks + 1) * P_ + col0 + m];
    c = WMMA_F32(a, b, c);
  }
  const int hi = (lane >> 4) * 8;
  for (int r = 0; r < 8; ++r)
    H[(row0 + r + hi) * P_ + col0 + m] = c[r] + b1[col0 + m];
}

// ---------------------------------------------------------------------------
// K3: fi = H @ Wi, fj = H @ Wj; center rows, emit per-row variances.
//   A[i,k] = fi[i,k] - mean_k fi[i,:]            ; sA2[i] = mean_k A^2
//   B[j,k] = fj[j,k] + bc1[k] - mean_k(fj+bc1)   ; sB2[j] = mean_k B^2
// One block = 16 rows. Waves 0-3 -> fi (64 cols each), waves 4-7 -> fj.
// ---------------------------------------------------------------------------
__global__ __launch_bounds__(256) void k_fifj(
    const float* __restrict__ H, const float* __restrict__ Wc1,
    const float* __restrict__ bc1,
    float* __restrict__ Amat, float* __restrict__ Bmat,
    float* __restrict__ sA2, float* __restrict__ sB2) {
  __shared__ __align__(16) float sH[16 * PADH];
  __shared__ __align__(16) float sFi[16 * PAD];
  __shared__ __align__(16) float sFj[16 * PAD];
  const int tid  = threadIdx.x;
  const int lane = tid & 31;
  const int wave = tid >> 5;
  const int row0 = blockIdx.x * 16;

  {  // stage 16x128 H block
    const int cc = (tid & 31) * 4, r0 = tid >> 5;
    for (int q = 0; q < 2; ++q) {
      int rr = r0 + q * 8;
      *(float4*)&sH[rr * PADH + cc] = *(const float4*)&H[(row0 + rr) * P_ + cc];
    }
  }
  __syncthreads();

  const int m  = lane & 15;
  const int ks = (lane >> 4) * 2;
  const bool isJ = (wave >= 4);
  const int cbase = (wave & 3) * 64;           // 4 tiles of 16 cols
  const int krow0 = isJ ? P_ : 0;              // Wj = Wc1[128:], Wi = Wc1[:128]
  v8f c0 = {}, c1 = {}, c2 = {}, c3 = {};
  for (int k0 = 0; k0 < P_; k0 += 4) {
    v2f a;
    a.x = sH[m * PADH + k0 + ks];
    a.y = sH[m * PADH + k0 + ks + 1];
    const float* wp = &Wc1[(krow0 + k0 + ks) * HD_ + cbase + m];
    v2f b0v; b0v.x = wp[0];  b0v.y = wp[HD_];
    v2f b1v; b1v.x = wp[16]; b1v.y = wp[HD_ + 16];
    v2f b2v; b2v.x = wp[32]; b2v.y = wp[HD_ + 32];
    v2f b3v; b3v.x = wp[48]; b3v.y = wp[HD_ + 48];
    c0 = WMMA_F32(a, b0v, c0);
    c1 = WMMA_F32(a, b1v, c1);
    c2 = WMMA_F32(a, b2v, c2);
    c3 = WMMA_F32(a, b3v, c3);
  }
  float* dst = isJ ? sFj : sFi;
  const int hi = (lane >> 4) * 8;
  for (int r = 0; r < 8; ++r) {
    int rr = r + hi;
    dst[rr * PAD + cbase + m]      = c0[r];
    dst[rr * PAD + cbase + 16 + m] = c1[r];
    dst[rr * PAD + cbase + 32 + m] = c2[r];
    dst[rr * PAD + cbase + 48 + m] = c3[r];
  }
  __syncthreads();

  // Row stats: row r handled by 16 threads; values held in registers.
  const int r  = tid >> 4;
  const int cg = tid & 15;
  {  // ---- fi -> A, sA2 ----
    const float4* prow = (const float4*)&sFi[r * PAD + cg * 16];
    float4 x0 = prow[0], x1 = prow[1], x2 = prow[2], x3 = prow[3];
    float s = hsum4(x0) + hsum4(x1) + hsum4(x2) + hsum4(x3);
    for (int o = 8; o >= 1; o >>= 1) s += __shfl_xor(s, o, 16);
    float mean = s * (1.0f / 256.0f);
    float4 dd[4] = {sub4(x0, mean), sub4(x1, mean), sub4(x2, mean), sub4(x3, mean)};
    float vs = dot4(dd[0], dd[0]) + dot4(dd[1], dd[1]) +
               dot4(dd[2], dd[2]) + dot4(dd[3], dd[3]);
    for (int o = 8; o >= 1; o >>= 1) vs += __shfl_xor(vs, o, 16);
#pragma unroll
    for (int q = 0; q < 4; ++q)
      ((float4*)&Amat[(row0 + r) * HD_])[cg * 4 + q] = dd[q];
    if (cg == 0) sA2[row0 + r] = vs * (1.0f / 256.0f);
  }
  {  // ---- fj + bc1 -> B, sB2 ----
    const float4* prow = (const float4*)&sFj[r * PAD + cg * 16];
    float4 x[4];
#pragma unroll
    for (int q = 0; q < 4; ++q) {
      float4 v = prow[q];
      float4 bb = ((const float4*)bc1)[cg * 4 + q];
      x[q] = make_float4(v.x + bb.x, v.y + bb.y, v.z + bb.z, v.w + bb.w);
    }
    float s = hsum4(x[0]) + hsum4(x[1]) + hsum4(x[2]) + hsum4(x[3]);
    for (int o = 8; o >= 1; o >>= 1) s += __shfl_xor(s, o, 16);
    float mean = s * (1.0f / 256.0f);
    float4 dd[4] = {sub4(x[0], mean), sub4(x[1], mean), sub4(x[2], mean), sub4(x[3], mean)};
    float vs = dot4(dd[0], dd[0]) + dot4(dd[1], dd[1]) +
               dot4(dd[2], dd[2]) + dot4(dd[3], dd[3]);
    for (int o = 8; o >= 1; o >>= 1) vs += __shfl_xor(vs, o, 16);
#pragma unroll
    for (int q = 0; q < 4; ++q)
      ((float4*)&Bmat[(row0 + r) * HD_])[cg * 4 + q] = dd[q];
    if (cg == 0) sB2[row0 + r] = vs * (1.0f / 256.0f);
  }
}

// ---------------------------------------------------------------------------
// K4 (dominant): 16x16 pair tile per block.
//   cross_ij = (1/256) sum_k A[i,k]*B[j,k]   via WMMA (8 waves split K=256)
//   var_ij   = sA2[i] + sB2[j] + 2*cross_ij  -> rstd
//   logit_ij = bc2 + sum_k mish((A+B)*rstd*gc[k] + bec[k]) * Wc2[k]
// Hot loop reads LDS in b128 quanta; pitch 260 keeps 16 jj-rows spread over
// all 64 banks.
// ---------------------------------------------------------------------------
__global__ __launch_bounds__(256) void k_pair(
    const float* __restrict__ Amat, const float* __restrict__ Bmat,
    const float* __restrict__ sA2, const float* __restrict__ sB2,
    const float* __restrict__ gc, const float* __restrict__ bec,
    const float* __restrict__ Wc2, const float* __restrict__ bc2,
    float* __restrict__ out) {
  __shared__ __align__(16) float sA[16 * PAD];
  __shared__ __align__(16) float sB[16 * PAD];
  __shared__ __align__(16) float sC[8 * 256];
  __shared__ float sVa[16], sVb[16];
  __shared__ __align__(16) float sG[256], sBe[256], sW2[256];
  const int tid  = threadIdx.x;
  const int lane = tid & 31;
  const int wave = tid >> 5;
  const int i0 = blockIdx.y * 16;
  const int j0 = blockIdx.x * 16;

  {  // stage A/B tiles, b128 coalesced
    const int cc = (tid & 63) * 4, r0 = tid >> 6;
    for (int q = 0; q < 4; ++q) {
      int rr = r0 + q * 4;
      *(float4*)&sA[rr * PAD + cc] = *(const float4*)&Amat[(i0 + rr) * HD_ + cc];
      *(float4*)&sB[rr * PAD + cc] = *(const float4*)&Bmat[(j0 + rr) * HD_ + cc];
    }
  }
  if (tid < 64) {
    ((float4*)sG)[tid]  = ((const float4*)gc)[tid];
    ((float4*)sBe)[tid] = ((const float4*)bec)[tid];
    ((float4*)sW2)[tid] = ((const float4*)Wc2)[tid];
  }
  if (tid < 16) { sVa[tid] = sA2[i0 + tid]; sVb[tid] = sB2[j0 + tid]; }
  __syncthreads();

  // Cross-term GEMM A(16x256) * B(16x256)^T: wave w handles K in [32w,32w+32)
  const int m  = lane & 15;
  const int ks = (lane >> 4) * 2;
  v8f c = {};
  for (int k0 = wave * 32; k0 < wave * 32 + 32; k0 += 4) {
    v2f a; a.x = sA[m * PAD + k0 + ks]; a.y = sA[m * PAD + k0 + ks + 1];
    v2f b; b.x = sB[m * PAD + k0 + ks]; b.y = sB[m * PAD + k0 + ks + 1];
    c = WMMA_F32(a, b, c);
  }
  const int hi = (lane >> 4) * 8;
  for (int r = 0; r < 8; ++r)
    sC[wave * 256 + (r + hi) * 16 + m] = c[r];
  __syncthreads();

  const int ii = tid >> 4;     // local i
  const int jj = tid & 15;     // local j
  float cross = 0.0f;
  for (int w = 0; w < 8; ++w) cross += sC[w * 256 + ii * 16 + jj];
  float var  = sVa[ii] + sVb[jj] + 2.0f * cross * (1.0f / 256.0f);
  float rstd = rsqrtf(var + 1e-5f);

  const float4* pA  = (const float4*)&sA[ii * PAD];
  const float4* pB  = (const float4*)&sB[jj * PAD];
  const float4* pG  = (const float4*)sG;
  const float4* pBe = (const float4*)sBe;
  const float4* pW  = (const float4*)sW2;
  float acc = 0.0f;
#pragma unroll 4
  for (int k4 = 0; k4 < HD_ / 4; ++k4) {
    float4 av = pA[k4], bv = pB[k4], gv = pG[k4], ev = pBe[k4], wv = pW[k4];
    float t0 = (av.x + bv.x) * rstd * gv.x + ev.x;
    float t1 = (av.y + bv.y) * rstd * gv.y + ev.y;
    float t2 = (av.z + bv.z) * rstd * gv.z + ev.z;
    float t3 = (av.w + bv.w) * rstd * gv.w + ev.w;
    acc += mishf(t0) * wv.x + mishf(t1) * wv.y + mishf(t2) * wv.z + mishf(t3) * wv.w;
  }
  out[(i0 + ii) * N_ + (j0 + jj)] = acc + bc2[0];
}

// ---------------------------------------------------------------------------
// K5: matrix_mask = ones -> fill tail of d_out with 1.0
// ---------------------------------------------------------------------------
__global__ void k_mask(float* __restrict__ out, int n) {
  int i = blockIdx.x * 256 + threadIdx.x;
  if (i < n) out[i] = 1.0f;
}

extern "C" void kernel_launch(void* const* d_in, const int* in_sizes, int n_in,
                              void* d_out, int out_size, void* d_ws, size_t ws_size,
                              hipStream_t stream) {
  const float* hf  = (const float*)d_in[0];   // (768,256)
  const float* W0  = (const float*)d_in[1];   // (256,256)
  const float* b0  = (const float*)d_in[2];   // (256,)
  const float* g0  = (const float*)d_in[3];   // (256,)
  const float* be0 = (const float*)d_in[4];   // (256,)
  const float* W1  = (const float*)d_in[5];   // (256,128)
  const float* b1  = (const float*)d_in[6];   // (128,)
  const float* Wc1 = (const float*)d_in[7];   // (256,256) = [Wi;Wj]
  const float* bc1 = (const float*)d_in[8];   // (256,)
  const float* gc  = (const float*)d_in[9];   // (256,)
  const float* bec = (const float*)d_in[10];  // (256,)
  const float* Wc2 = (const float*)d_in[11];  // (256,1)
  const float* bc2 = (const float*)d_in[12];  // (1,)

  float* ws   = (float*)d_ws;
  float* H1   = ws;                       // 768*256
  float* H    = H1 + N_ * HD_;            // 768*128
  float* Amat = H + N_ * P_;              // 768*256
  float* Bmat = Amat + N_ * HD_;          // 768*256
  float* vA   = Bmat + N_ * HD_;          // 768
  float* vB   = vA + N_;                  // 768

  k_gemm1_ln_mish<<<N_ / 16, 256, 0, stream>>>(hf, W0, b0, g0, be0, H1);
  k_gemm2<<<N_ / 16, 256, 0, stream>>>(H1, W1, b1, H);
  k_fifj<<<N_ / 16, 256, 0, stream>>>(H, Wc1, bc1, Amat, Bmat, vA, vB);
  dim3 grid(N_ / 16, N_ / 16);
  k_pair<<<grid, 256, 0, stream>>>(Amat, Bmat, vA, vB, gc, bec, Wc2, bc2,
                                   (float*)d_out);
  int total = N_ * N_;
  int rem = out_size - total;             // mask region (bool ones)
  if (rem > 0)
    k_mask<<<(rem + 255) / 256, 256, 0, stream>>>((float*)d_out + total, rem);
}